// DiceBoundaryLoss_85194971284149
// MI455X (gfx1250) — compile-verified
//
#include <hip/hip_runtime.h>
#include <math.h>

#define Bn 8
#define Hn 256
#define Wn 256
#define NPIX (Bn * Hn * Wn)
#define EDT_INF 1.0e10f
#define EPSV 1e-6f

typedef float v2f __attribute__((ext_vector_type(2)));
typedef float v8f __attribute__((ext_vector_type(8)));

// ---------------------------------------------------------------------------
// Wave32 sum via V_WMMA_F32_16X16X4_F32.
// A (16x4 f32, 2 VGPRs/lane): lane l<16 holds (M=l,K=0),(M=l,K=1);
// lane l>=16 holds (M=l-16,K=2),(M=l-16,K=3).  Put v in component 0, 0 in
// component 1, B = all-ones -> D[m,n] = v[m] + v[m+16] for every column n.
// Lane n holds D[0..7,n] (c[0..7]); lane n+16 holds D[8..15,n].
// total = (sum of own 8 comps) + partner lane via shfl_xor(16).
// EXEC is all-ones at every call site (full 256-thread blocks, no divergence).
// ---------------------------------------------------------------------------
__device__ __forceinline__ float wave_sum32(float v) {
    v2f a; a.x = v;    a.y = 0.0f;
    v2f b; b.x = 1.0f; b.y = 1.0f;
    v8f c = {0.0f, 0.0f, 0.0f, 0.0f, 0.0f, 0.0f, 0.0f, 0.0f};
    c = __builtin_amdgcn_wmma_f32_16x16x4_f32(
            /*neg_a=*/false, a, /*neg_b=*/false, b,
            /*c_mod=*/(short)0, c, /*reuse_a=*/false, /*reuse_b=*/false);
    float s = c[0] + c[1] + c[2] + c[3] + c[4] + c[5] + c[6] + c[7];
    s += __shfl_xor(s, 16, 32);
    return s;
}

// Deterministic 256-thread block sum (8 waves). Result valid in thread 0.
__device__ __forceinline__ float block_sum256(float v, float* lds8) {
    float w = wave_sum32(v);
    int wid = threadIdx.x >> 5;
    if ((threadIdx.x & 31) == 0) lds8[wid] = w;
    __syncthreads();
    float r = 0.0f;
    if (threadIdx.x == 0) {
        for (int i = 0; i < 8; ++i) r += lds8[i];
    }
    __syncthreads();   // make lds8 reusable by a subsequent call
    return r;
}

__device__ __forceinline__ float sigmoidf_dev(float x) {
    return 1.0f / (1.0f + expf(-x));
}

// ---------------------------------------------------------------------------
// Pass 1: min-plus along W for both masks.
//   f_out[j] = (t<0.5) ? INF : 0   (EDT of 1-t)
//   f_in [j] = (t>0.5) ? INF : 0   (EDT of t)
// One block per image row (b*Hn + y); thread = x.  Coalesced in/out.
// ---------------------------------------------------------------------------
__global__ void edt_pass_w(const float* __restrict__ target,
                           float* __restrict__ g_out,
                           float* __restrict__ g_in) {
    __shared__ float f_o[Wn];
    __shared__ float f_i[Wn];
    const int row = blockIdx.x;          // b*Hn + y
    const int x   = threadIdx.x;
    float t = target[row * Wn + x];
    f_o[x] = (t < 0.5f) ? EDT_INF : 0.0f;
    f_i[x] = (t > 0.5f) ? EDT_INF : 0.0f;
    __syncthreads();

    float mo = 3.0e38f, mi = 3.0e38f;
#pragma unroll 8
    for (int j = 0; j < Wn; ++j) {
        float d  = (float)(x - j);
        float d2 = d * d;
        mo = fminf(mo, f_o[j] + d2);
        mi = fminf(mi, f_i[j] + d2);
    }
    g_out[row * Wn + x] = mo;
    g_in [row * Wn + x] = mi;
}

// ---------------------------------------------------------------------------
// Pass 2: min-plus along H for both masks + boundary-loss accumulation.
// Block = (batch b, 16-column tile x0..x0+15); 256 threads.
// Stages 256x16 tiles of g_out/g_in in LDS (2 x 16 KB), then each thread
// computes 16 outputs: dist = sqrt(h_out) + sqrt(h_in); acc += sigmoid(p)*dist.
// Per-block partial written via WMMA reduction (deterministic).
// ---------------------------------------------------------------------------
__global__ void edt_pass_h_bloss(const float* __restrict__ pred,
                                 const int*   __restrict__ from_logits,
                                 const float* __restrict__ g_out,
                                 const float* __restrict__ g_in,
                                 float* __restrict__ pd_partial) {
    __shared__ float co[Hn][16];
    __shared__ float ci[Hn][16];
    __shared__ float wsum[8];

    const int b   = blockIdx.x >> 4;       // 16 x-tiles per batch
    const int x0  = (blockIdx.x & 15) * 16;
    const int tid = threadIdx.x;
    const int lx  = tid & 15;              // column within tile
    const int ly  = tid >> 4;              // 0..15

    // Load both 256x16 column tiles (16 consecutive lanes -> 64B segments).
#pragma unroll 4
    for (int ld = 0; ld < 16; ++ld) {
        int y    = ld * 16 + ly;
        int gidx = (b * Hn + y) * Wn + x0 + lx;
        co[y][lx] = g_out[gidx];
        ci[y][lx] = g_in [gidx];
    }
    __syncthreads();

    const bool logits = (from_logits[0] != 0);
    float acc = 0.0f;

    for (int k = 0; k < 16; ++k) {
        int y = k * 16 + ly;
        float mo = 3.0e38f, mi = 3.0e38f;
#pragma unroll 8
        for (int j = 0; j < Hn; ++j) {
            float d  = (float)(y - j);
            float d2 = d * d;
            mo = fminf(mo, co[j][lx] + d2);
            mi = fminf(mi, ci[j][lx] + d2);
        }
        float dist = sqrtf(mo) + sqrtf(mi);
        float p = pred[(b * Hn + y) * Wn + x0 + lx];
        if (logits) p = sigmoidf_dev(p);
        acc += p * dist;
    }

    float tot = block_sum256(acc, wsum);
    if (tid == 0) pd_partial[blockIdx.x] = tot;
}

// ---------------------------------------------------------------------------
// Dice sums: per-block partials of (p*t, p*p, t*t). Block covers 4096
// consecutive elements; 16 blocks per batch, so per-batch sums stay separable.
// ---------------------------------------------------------------------------
__global__ void dice_partials(const float* __restrict__ pred,
                              const float* __restrict__ target,
                              const int*   __restrict__ from_logits,
                              float* __restrict__ dice_partial /* [128][3] */) {
    __shared__ float wsum[8];
    const bool logits = (from_logits[0] != 0);
    const int  base   = blockIdx.x * 4096;

    float apt = 0.0f, app = 0.0f, att = 0.0f;
#pragma unroll 4
    for (int it = 0; it < 16; ++it) {
        int idx = base + it * 256 + threadIdx.x;
        float p = pred[idx];
        if (logits) p = sigmoidf_dev(p);
        float t = target[idx];
        apt += p * t;
        app += p * p;
        att += t * t;
    }
    float spt = block_sum256(apt, wsum);
    float spp = block_sum256(app, wsum);
    float stt = block_sum256(att, wsum);
    if (threadIdx.x == 0) {
        dice_partial[blockIdx.x * 3 + 0] = spt;
        dice_partial[blockIdx.x * 3 + 1] = spp;
        dice_partial[blockIdx.x * 3 + 2] = stt;
    }
}

// ---------------------------------------------------------------------------
// Final scalar combine (single thread, fixed order -> deterministic).
// ---------------------------------------------------------------------------
__global__ void finalize(const float* __restrict__ dice_partial, // [128][3]
                         const float* __restrict__ pd_partial,   // [128]
                         float* __restrict__ out) {
    if (threadIdx.x != 0 || blockIdx.x != 0) return;
    float dsum = 0.0f;
    for (int b = 0; b < Bn; ++b) {
        float pt = 0.0f, pp = 0.0f, tt = 0.0f;
        for (int k = 0; k < 16; ++k) {
            int blk = b * 16 + k;
            pt += dice_partial[blk * 3 + 0];
            pp += dice_partial[blk * 3 + 1];
            tt += dice_partial[blk * 3 + 2];
        }
        dsum += 1.0f - (2.0f * pt + EPSV) / (pp + tt + EPSV);
    }
    float d_loss = dsum / (float)Bn;

    float pd = 0.0f;
    for (int i = 0; i < 128; ++i) pd += pd_partial[i];
    float b_loss = pd / (float)NPIX;

    out[0] = d_loss + b_loss;
}

// ---------------------------------------------------------------------------
// Launch
// ---------------------------------------------------------------------------
extern "C" void kernel_launch(void* const* d_in, const int* in_sizes, int n_in,
                              void* d_out, int out_size, void* d_ws, size_t ws_size,
                              hipStream_t stream) {
    const float* pred   = (const float*)d_in[0];   // [B,1,H,W] fp32
    const float* target = (const float*)d_in[1];   // [B,1,H,W] fp32
    const int*   fl     = (const int*)  d_in[2];   // scalar flag
    float*       out    = (float*)d_out;

    float* ws           = (float*)d_ws;
    float* g_out        = ws;                      // NPIX floats
    float* g_in         = ws + NPIX;               // NPIX floats
    float* pd_partial   = ws + 2 * NPIX;           // 128 floats
    float* dice_partial = ws + 2 * NPIX + 128;     // 128*3 floats

    edt_pass_w     <<<Bn * Hn, Wn,  0, stream>>>(target, g_out, g_in);
    dice_partials  <<<128,     256, 0, stream>>>(pred, target, fl, dice_partial);
    edt_pass_h_bloss<<<128,    256, 0, stream>>>(pred, fl, g_out, g_in, pd_partial);
    finalize       <<<1,       32,  0, stream>>>(dice_partial, pd_partial, out);
}